// v_ProjectLayer_20615843021366
// MI455X (gfx1250) — compile-verified
//
#include <hip/hip_runtime.h>
#include <hip/hip_bf16.h>

// CDNA5 / gfx1250. wave32. One workgroup (256 thr = 8 waves) per coarse site.
// Memory-bound streaming kernel: ~117MB read once, 48KB written.
// WMMA (V_WMMA_F32_16X16X4_F32, ones-A trick) performs the final 16-row
// column-sum of the block pooling. __builtin_prefetch -> global_prefetch_b8.

typedef float v2f __attribute__((ext_vector_type(2)));
typedef float v8f __attribute__((ext_vector_type(8)));

#define NSITES 131072   // 16*16*16*32
#define LDS_STRIDE 36   // floats per row: 32 data cols + 4 pad (bank skew, 16B aligned)

__device__ __forceinline__ void cfma(float2& acc, const float2 a, const float2 b) {
    // acc += a * b (complex)
    acc.x = fmaf(a.x, b.x, acc.x);
    acc.x = fmaf(-a.y, b.y, acc.x);
    acc.y = fmaf(a.x, b.y, acc.y);
    acc.y = fmaf(a.y, b.x, acc.y);
}

__global__ void __launch_bounds__(256)
v_project_pool_kernel(const float* __restrict__ fea,
                      const float* __restrict__ wts,
                      const float* __restrict__ gfl,
                      float* __restrict__ out) {
    __shared__ float lds[256 * LDS_STRIDE];

    const int tid = threadIdx.x;       // 0..255
    const int blk = blockIdx.x;        // 0..511 == coarse flat index

    // Coarse coords: flat = ((cx*4+cy)*4+cz)*8+ct
    const int ct = blk & 7;
    const int cz = (blk >> 3) & 3;
    const int cy = (blk >> 5) & 3;
    const int cx = (blk >> 7) & 3;
    // Fine offset within 4x4x4x4 block
    const int bt = tid & 3;
    const int bz = (tid >> 2) & 3;
    const int by = (tid >> 4) & 3;
    const int bx = (tid >> 6) & 3;

    const int x = cx * 4 + bx, y = cy * 4 + by, z = cz * 4 + bz, t = ct * 4 + bt;
    const int site = ((x * 16 + y) * 16 + z) * 32 + t;   // 0..131071

    // ---- load fea for this site: 12 complex = 96B contiguous, 16B aligned ----
    float2 f[4][3];
    {
        const float4* fp = reinterpret_cast<const float4*>(fea + (size_t)site * 24);
        #pragma unroll
        for (int q = 0; q < 6; ++q) {
            float4 v = fp[q];
            int e0 = 2 * q, e1 = 2 * q + 1;
            f[e0 / 3][e0 % 3] = make_float2(v.x, v.y);
            f[e1 / 3][e1 % 3] = make_float2(v.z, v.w);
        }
    }

    float2 st[4][3];
    #pragma unroll
    for (int i = 0; i < 4; ++i)
        #pragma unroll
        for (int a = 0; a < 3; ++a) st[i][a] = make_float2(0.f, 0.f);

    #pragma unroll
    for (int p = 0; p < 4; ++p) {
        const size_t psite = (size_t)p * NSITES + site;

        // Prefetch next path's lines while we compute this one (global_prefetch_b8)
        if (p < 3) {
            const size_t nsite = psite + NSITES;
            __builtin_prefetch(wts + nsite * 32, 0, 0);
            __builtin_prefetch(gfl + nsite * 18, 0, 0);
        }

        // weights: 16 complex = 128B contiguous, 16B aligned
        float2 W[4][4];
        {
            const float4* wp = reinterpret_cast<const float4*>(wts + psite * 32);
            #pragma unroll
            for (int q = 0; q < 8; ++q) {
                float4 v = wp[q];
                int e0 = 2 * q, e1 = 2 * q + 1;
                W[e0 / 4][e0 % 4] = make_float2(v.x, v.y);
                W[e1 / 4][e1 % 4] = make_float2(v.z, v.w);
            }
        }
        // gauge: 9 complex = 72B contiguous, only 8B aligned -> float2 loads
        float2 U[3][3];
        {
            const float2* up = reinterpret_cast<const float2*>(gfl + psite * 18);
            #pragma unroll
            for (int q = 0; q < 9; ++q) U[q / 3][q % 3] = up[q];
        }

        // gt[s][a] = sum_b U[a][b] * f[s][b]
        float2 gt[4][3];
        #pragma unroll
        for (int s = 0; s < 4; ++s)
            #pragma unroll
            for (int a = 0; a < 3; ++a) {
                float2 acc = make_float2(0.f, 0.f);
                #pragma unroll
                for (int b = 0; b < 3; ++b) cfma(acc, U[a][b], f[s][b]);
                gt[s][a] = acc;
            }
        // st[i][a] += sum_j W[i][j] * gt[j][a]
        #pragma unroll
        for (int i = 0; i < 4; ++i)
            #pragma unroll
            for (int a = 0; a < 3; ++a)
                #pragma unroll
                for (int j = 0; j < 4; ++j) cfma(st[i][a], W[i][j], gt[j][a]);
    }

    // ---- write per-thread result to LDS: 24 floats + zero pad cols 24..31 ----
    {
        float* row = &lds[tid * LDS_STRIDE];
        #pragma unroll
        for (int i = 0; i < 4; ++i)
            #pragma unroll
            for (int a = 0; a < 3; ++a) {
                row[(i * 3 + a) * 2 + 0] = st[i][a].x;
                row[(i * 3 + a) * 2 + 1] = st[i][a].y;
            }
        #pragma unroll
        for (int c = 24; c < 32; ++c) row[c] = 0.f;
    }
    __syncthreads();

    // ---- LDS tree: 256 rows -> 16 rows (cols 0..31) ----
    #pragma unroll
    for (int step = 128; step >= 16; step >>= 1) {
        if (tid < step) {
            float4* a4 = reinterpret_cast<float4*>(&lds[tid * LDS_STRIDE]);
            const float4* b4 = reinterpret_cast<const float4*>(&lds[(tid + step) * LDS_STRIDE]);
            #pragma unroll
            for (int c = 0; c < 8; ++c) {
                float4 va = a4[c];
                float4 vb = b4[c];
                va.x += vb.x; va.y += vb.y; va.z += vb.z; va.w += vb.w;
                a4[c] = va;
            }
        }
        __syncthreads();
    }

    // ---- wave 0: WMMA column-sum of the remaining 16x32 block ----
    // A = all-ones 16x4  =>  D[m][n] = sum_k B[k][n] for every m, so the
    // result is invariant to the exact K/M operand permutation; only the
    // lane -> N = lane%16 striping of B and D is relied upon.
    if (tid < 32) {
        const int lane = tid;
        const int n = lane & 15;
        const int half = lane >> 4;

        v2f amat;
        amat[0] = 1.0f;
        amat[1] = 1.0f;

        v8f acc0 = {};
        v8f acc1 = {};
        #pragma unroll
        for (int k = 0; k < 4; ++k) {
            const int r0 = 4 * k + 2 * half;   // this lane-half's two K rows
            v2f b0, b1;
            b0[0] = lds[(r0 + 0) * LDS_STRIDE + n];
            b0[1] = lds[(r0 + 1) * LDS_STRIDE + n];
            b1[0] = lds[(r0 + 0) * LDS_STRIDE + 16 + n];
            b1[1] = lds[(r0 + 1) * LDS_STRIDE + 16 + n];
            acc0 = __builtin_amdgcn_wmma_f32_16x16x4_f32(false, amat, false, b0,
                                                         (short)0, acc0, false, false);
            acc1 = __builtin_amdgcn_wmma_f32_16x16x4_f32(false, amat, false, b1,
                                                         (short)0, acc1, false, false);
        }

        float* op = out + (size_t)blk * 24;
        if (lane < 16) op[n] = acc0[0];        // columns 0..15
        if (lane < 8)  op[16 + n] = acc1[0];   // columns 16..23 (24..31 are pad)
    }
}

extern "C" void kernel_launch(void* const* d_in, const int* in_sizes, int n_in,
                              void* d_out, int out_size, void* d_ws, size_t ws_size,
                              hipStream_t stream) {
    (void)in_sizes; (void)n_in; (void)out_size; (void)d_ws; (void)ws_size;
    const float* fea = (const float*)d_in[0];   // complex64 [1,16,16,16,32,4,3]
    const float* wts = (const float*)d_in[1];   // complex64 [4,16,16,16,32,4,4]
    const float* gfl = (const float*)d_in[2];   // complex64 [4,16,16,16,32,3,3]
    float* out = (float*)d_out;                 // complex64 [1,4,4,4,8,4,3]

    v_project_pool_kernel<<<dim3(512), dim3(256), 0, stream>>>(fea, wts, gfl, out);
}